// APPNP_ModelBen1_45792941310033
// MI455X (gfx1250) — compile-verified
//
#include <hip/hip_runtime.h>
#include <hip/hip_bf16.h>

typedef __attribute__((ext_vector_type(2))) float v2f;
typedef __attribute__((ext_vector_type(8))) float v8f;

#define N_NODES   100000
#define N_EDGES   1600000
#define IN_DIM    512
#define OUT_DIM   32
#define K_STEPS   10
#define ALPHA     0.1f

// ---------------------------------------------------------------------------
// h = x @ W^T + b  via V_WMMA_F32_16X16X4_F32.
// One wave per 16-row tile of h; two 16-col N-tiles (OUT_DIM=32).
// A (16x4 f32): lane m=lane&15 holds row m; VGPR0/1 = K {0,1} (lanes 0-15)
//               or K {2,3} (lanes 16-31)  -> per-lane float2 x[row][kk..kk+1].
// B (4x16 f32): lane n=lane&15 holds col n; same K split -> per-lane float2
//               W[n][kk..kk+1], staged transposed in LDS as [kpair][n] float2.
// C (16x16 f32): VGPR r: lanes0-15 M=r, lanes16-31 M=r+8.
// ---------------------------------------------------------------------------
__launch_bounds__(320)
__global__ void appnp_proj_wmma(const float* __restrict__ x,
                                const float* __restrict__ W,
                                const float* __restrict__ bias,
                                float* __restrict__ h) {
  __shared__ v2f Wlds[(IN_DIM / 2) * OUT_DIM];  // 8192 * 8B = 64 KB

  const int tid = threadIdx.x;
  // Stage W transposed: t = n*256 + kp  -> coalesced 8B global reads.
  for (int t = tid; t < OUT_DIM * (IN_DIM / 2); t += 320) {
    const int n  = t >> 8;     // 0..31
    const int kp = t & 255;    // k-pair 0..255
    v2f v;
    v.x = W[(size_t)n * IN_DIM + 2 * kp + 0];
    v.y = W[(size_t)n * IN_DIM + 2 * kp + 1];
    Wlds[kp * OUT_DIM + n] = v;
  }
  __syncthreads();

  const int wave    = tid >> 5;
  const int lane    = tid & 31;
  const int tile    = blockIdx.x * 10 + wave;   // 625*10 = 6250 tiles exactly
  const int rowbase = tile * 16;
  const int m       = lane & 15;
  const int khalf   = lane >> 4;                // 0 or 1

  const float* xrow = x + (size_t)(rowbase + m) * IN_DIM;

  v8f c0 = {};
  v8f c1 = {};
  for (int k0 = 0; k0 < IN_DIM; k0 += 4) {
    const int kk = k0 + 2 * khalf;
    const v2f a  = *(const v2f*)(xrow + kk);
    const v2f b0 = Wlds[(kk >> 1) * OUT_DIM + m];
    const v2f b1 = Wlds[(kk >> 1) * OUT_DIM + 16 + m];
    c0 = __builtin_amdgcn_wmma_f32_16x16x4_f32(false, a, false, b0,
                                               (short)0, c0, false, false);
    c1 = __builtin_amdgcn_wmma_f32_16x16x4_f32(false, a, false, b1,
                                               (short)0, c1, false, false);
  }

  const float bias0 = bias[m];
  const float bias1 = bias[16 + m];
  const int rb = rowbase + khalf * 8;
  for (int r = 0; r < 8; ++r) {
    float* hr = h + (size_t)(rb + r) * OUT_DIM;
    hr[m]      = c0[r] + bias0;
    hr[16 + m] = c1[r] + bias1;
  }
}

// ---------------------------------------------------------------------------
// Degree / normalization / edge-weight kernels
// ---------------------------------------------------------------------------
__global__ void appnp_zero_f32(float* __restrict__ p, int n) {
  int i = blockIdx.x * blockDim.x + threadIdx.x;
  if (i < n) p[i] = 0.0f;
}

__global__ void appnp_degree(const int* __restrict__ ei, float* __restrict__ deg) {
  int e = blockIdx.x * blockDim.x + threadIdx.x;
  if (e < N_EDGES) {
    int dst = ei[N_EDGES + e];
    atomicAdd(&deg[dst], 1.0f);
  }
}

// deg <- 1/sqrt(deg + 1)   (+1 = self loop; always > 0)
__global__ void appnp_inv_sqrt(float* __restrict__ deg) {
  int i = blockIdx.x * blockDim.x + threadIdx.x;
  if (i < N_NODES) deg[i] = rsqrtf(deg[i] + 1.0f);
}

__global__ void appnp_edge_w(const int* __restrict__ ei,
                             const float* __restrict__ dis,
                             float* __restrict__ w) {
  int e = blockIdx.x * blockDim.x + threadIdx.x;
  if (e < N_EDGES) {
    int src = ei[e];
    int dst = ei[N_EDGES + e];
    w[e] = dis[src] * dis[dst];
  }
}

// ---------------------------------------------------------------------------
// Propagation step, split in two:
//   init:  z_next = ALPHA*h + (1-ALPHA)*dis[i]^2 * z_cur     (self-loop fused)
//   edges: z_next[dst] += (1-ALPHA)*w[e] * z_cur[src]        (f32 atomics)
// ---------------------------------------------------------------------------
__global__ void appnp_step_init(const float* __restrict__ h,
                                const float* __restrict__ z_cur,
                                const float* __restrict__ dis,
                                float* __restrict__ z_next) {
  int i = blockIdx.x * blockDim.x + threadIdx.x;
  if (i < N_NODES * OUT_DIM) {
    int node = i >> 5;  // OUT_DIM == 32
    float d = dis[node];
    z_next[i] = ALPHA * h[i] + (1.0f - ALPHA) * d * d * z_cur[i];
  }
}

__global__ void appnp_step_edges(const int* __restrict__ ei,
                                 const float* __restrict__ w,
                                 const float* __restrict__ z_cur,
                                 float* __restrict__ z_next) {
  int gid = blockIdx.x * blockDim.x + threadIdx.x;
  if (gid < N_EDGES * 8) {
    int e  = gid >> 3;
    int c0 = (gid & 7) * 4;
    int src = ei[e];
    int dst = ei[N_EDGES + e];
    float coef = (1.0f - ALPHA) * w[e];
    const float4 v = *(const float4*)(z_cur + (size_t)src * OUT_DIM + c0);
    float* outp = z_next + (size_t)dst * OUT_DIM + c0;
    atomicAdd(outp + 0, coef * v.x);
    atomicAdd(outp + 1, coef * v.y);
    atomicAdd(outp + 2, coef * v.z);
    atomicAdd(outp + 3, coef * v.w);
  }
}

// ---------------------------------------------------------------------------
extern "C" void kernel_launch(void* const* d_in, const int* in_sizes, int n_in,
                              void* d_out, int out_size, void* d_ws, size_t ws_size,
                              hipStream_t stream) {
  const float* x  = (const float*)d_in[0];   // [N_NODES, IN_DIM]
  const float* W  = (const float*)d_in[1];   // [OUT_DIM, IN_DIM]
  const float* b  = (const float*)d_in[2];   // [OUT_DIM]
  const int*   ei = (const int*)d_in[3];     // [2, N_EDGES] (src row, dst row)

  float* out = (float*)d_out;                // [N_NODES, OUT_DIM]

  // Workspace layout (floats): h | zA | deg/dis | w  ~= 32.5 MB
  float* ws  = (float*)d_ws;
  float* h   = ws;                                   // 3,200,000
  float* zA  = ws + (size_t)N_NODES * OUT_DIM;       // 3,200,000
  float* deg = zA + (size_t)N_NODES * OUT_DIM;       //   100,000
  float* w   = deg + N_NODES;                        // 1,600,000

  // 1) Projection (WMMA f32): h = x @ W^T + b
  appnp_proj_wmma<<<625, 320, 0, stream>>>(x, W, b, h);

  // 2) gcn_norm: degrees (with self loop) -> 1/sqrt -> per-edge weight
  appnp_zero_f32<<<(N_NODES + 255) / 256, 256, 0, stream>>>(deg, N_NODES);
  appnp_degree<<<(N_EDGES + 255) / 256, 256, 0, stream>>>(ei, deg);
  appnp_inv_sqrt<<<(N_NODES + 255) / 256, 256, 0, stream>>>(deg);
  appnp_edge_w<<<(N_EDGES + 255) / 256, 256, 0, stream>>>(ei, deg, w);

  // 3) K propagation steps; ping-pong zA <-> d_out so step 10 lands in d_out.
  const float* z_cur = h;
  for (int s = 0; s < K_STEPS; ++s) {
    float* z_next = (s & 1) ? out : zA;   // s = 9 (last) is odd -> out
    appnp_step_init<<<(N_NODES * OUT_DIM + 255) / 256, 256, 0, stream>>>(
        h, z_cur, deg, z_next);
    appnp_step_edges<<<(N_EDGES * 8 + 255) / 256, 256, 0, stream>>>(
        ei, w, z_cur, z_next);
    z_cur = z_next;
  }
}